// MoSAN_20091857010802
// MI455X (gfx1250) — compile-verified
//
#include <hip/hip_runtime.h>

#define D 64
#define MEMB 8
#define GPB 128            // groups per block
#define THREADS 256
#define US 68              // padded U/g LDS row stride (floats): bank = 4*row + col, conflict-free

typedef __attribute__((ext_vector_type(2))) float v2f;
typedef __attribute__((ext_vector_type(8))) float v8f;

// ---------------------------------------------------------------------------
// Kernel A: fold W_v, W_o, b_v, b_o into C[d][f] = sum_e W_v[e,d]*W_o[f,e]
// and cvec[f] = 8*(b_o[f] + sum_e W_o[f,e]*b_v[e]).  Tiny (0.5 MFLOP).
// ---------------------------------------------------------------------------
__global__ void __launch_bounds__(THREADS)
mosan_prep_kernel(const float* __restrict__ W_v, const float* __restrict__ b_v,
                  const float* __restrict__ W_o, const float* __restrict__ b_o,
                  float* __restrict__ Cmat, float* __restrict__ cvec) {
  const int i = blockIdx.x * blockDim.x + threadIdx.x;
  if (i < D * D) {
    const int f = i & (D - 1);
    const int d = i >> 6;
    float s = 0.f;
    #pragma unroll 8
    for (int e = 0; e < D; ++e)
      s = __builtin_fmaf(W_o[f * D + e], W_v[e * D + d], s);
    Cmat[d * D + f] = s;            // row-major C[d][f], coalesced store
  }
  if (i < D) {
    float s = b_o[i];
    for (int e = 0; e < D; ++e)
      s = __builtin_fmaf(W_o[i * D + e], b_v[e], s);
    cvec[i] = 8.0f * s;
  }
}

// ---------------------------------------------------------------------------
// Kernel B: per block of 128 groups:
//   1) gather-sum 8 member rows -> U[128][64] in LDS
//   2) per wave: g_tile(16x64) = U_tile(16x64) @ C(64x64) via v_wmma_f32_16x16x4_f32
//      (C pre-swizzled fragment-major: each B frag = one aligned ds_load_b64)
//   3) y[b] = dot(g[b]+cvec, item_emb[item_inputs[b]])
// ---------------------------------------------------------------------------
__global__ void __launch_bounds__(THREADS)
mosan_main_kernel(const float* __restrict__ user_emb,
                  const float* __restrict__ item_emb,
                  const float* __restrict__ Cmat,
                  const float* __restrict__ cvec,
                  const int* __restrict__ members,
                  const int* __restrict__ item_inputs,
                  float* __restrict__ out, int B) {
  __shared__ float Ul[GPB * US];   // 34816 B : U, later reused for g
  __shared__ float Cl[D * D];      // 16384 B : fragment-major swizzle [kk][n][h][c][j]
  __shared__ float cv[D];

  const int tid = threadIdx.x;
  const int b0  = blockIdx.x * GPB;

  // ---- stage C into LDS, swizzled so each WMMA B-fragment is contiguous ----
  // element C[k][f]:  k = kk*4 + h*2 + j   (h = lane-half, j = frag slot)
  //                   f = n*16 + c
  // flat = (((kk*4 + n)*2 + h)*16 + c)*2 + j
  for (int i = tid; i < D * D; i += THREADS) {
    const int k = i >> 6, f = i & (D - 1);   // Cmat is row-major C[k][f]
    const int kk = k >> 2, h = (k >> 1) & 1, j = k & 1;
    const int n = f >> 4, c = f & 15;
    Cl[((((kk * 4 + n) * 2 + h) * 16 + c) << 1) + j] = Cmat[i];
  }
  if (tid < D) cv[tid] = cvec[tid];

  // ---- gather + sum member embeddings: 16 threads per group, float4 lanes ----
  for (int slot = tid; slot < GPB * 16; slot += THREADS) {
    const int g   = slot >> 4;
    const int seg = slot & 15;                 // float4 segment (4 cols)
    const int b   = b0 + g;
    float4 acc = make_float4(0.f, 0.f, 0.f, 0.f);
    if (b < B) {
      const int* mrow = members + b * MEMB;
      #pragma unroll
      for (int m = 0; m < MEMB; ++m) {
        const float4 e =
            *(const float4*)(user_emb + (size_t)mrow[m] * D + seg * 4);
        acc.x += e.x; acc.y += e.y; acc.z += e.z; acc.w += e.w;
      }
    }
    float* up = Ul + g * US + seg * 4;
    up[0] = acc.x; up[1] = acc.y; up[2] = acc.z; up[3] = acc.w;
  }
  __syncthreads();

  // ---- fp32 WMMA: each wave owns 16 group-rows ----
  const int wave   = tid >> 5;
  const int lane   = tid & 31;
  const int gr     = wave * 16;                // first group-row of tile
  const int mrow16 = lane & 15;
  const int hsel   = lane >> 4;                // lane half
  const int khalf  = hsel << 1;                // 0 or 2

  const float* arow = Ul + (gr + mrow16) * US + khalf;  // 8B-aligned (272*r + 8h)
  const int    coff = hsel * 32 + (mrow16 << 1);

  v8f acc[4] = {v8f{}, v8f{}, v8f{}, v8f{}};
  #pragma unroll
  for (int kk = 0; kk < 16; ++kk) {            // K = 64 in steps of 4
    const v2f a = *(const v2f*)(arow + kk * 4);          // A 16x4 f32 fragment
    #pragma unroll
    for (int n = 0; n < 4; ++n) {              // 4 N-tiles of 16 cols
      const v2f bf = *(const v2f*)(Cl + (kk * 4 + n) * 64 + coff);  // one b64
      acc[n] = __builtin_amdgcn_wmma_f32_16x16x4_f32(
          /*neg_a=*/false, a, /*neg_b=*/false, bf,
          /*c_mod=*/(short)0, acc[n], /*reuse_a=*/false, /*reuse_b=*/false);
    }
  }

  // ---- spill g tiles back to LDS (reuse U region; per-wave rows, in-order DS) ----
  #pragma unroll
  for (int j = 0; j < 8; ++j) {
    const int row = gr + ((lane < 16) ? j : (j + 8)); // C/D layout: vgpr j -> M=j / M=j+8
    #pragma unroll
    for (int n = 0; n < 4; ++n)
      Ul[row * US + n * 16 + mrow16] = acc[n][j];
  }
  __syncthreads();

  // ---- final dot with item embedding ----
  if (tid < GPB) {
    const int b = b0 + tid;
    if (b < B) {
      const float* it = item_emb + (size_t)item_inputs[b] * D;
      const float* gp = Ul + tid * US;
      float s = 0.f;
      #pragma unroll 16
      for (int f = 0; f < D; ++f)
        s = __builtin_fmaf(gp[f] + cv[f], it[f], s);
      out[b] = s;
    }
  }
}

extern "C" void kernel_launch(void* const* d_in, const int* in_sizes, int n_in,
                              void* d_out, int out_size, void* d_ws, size_t ws_size,
                              hipStream_t stream) {
  const float* user_emb    = (const float*)d_in[0];
  // d_in[1] quser_emb, d_in[3..6] W_q/b_q/W_k/b_k, d_in[11..12] w_t/b_t:
  // provably unused (attention weights are data-independent here).
  const float* item_emb    = (const float*)d_in[2];
  const float* W_v         = (const float*)d_in[7];
  const float* b_v         = (const float*)d_in[8];
  const float* W_o         = (const float*)d_in[9];
  const float* b_o         = (const float*)d_in[10];
  const int*   members     = (const int*)d_in[13];
  const int*   item_inputs = (const int*)d_in[14];
  float*       out         = (float*)d_out;

  const int B = in_sizes[13] / MEMB;

  float* Cmat = (float*)d_ws;          // 4096 floats
  float* cvec = Cmat + D * D;          // 64 floats

  mosan_prep_kernel<<<(D * D + THREADS - 1) / THREADS, THREADS, 0, stream>>>(
      W_v, b_v, W_o, b_o, Cmat, cvec);

  const int nblocks = (B + GPB - 1) / GPB;
  mosan_main_kernel<<<nblocks, THREADS, 0, stream>>>(
      user_emb, item_emb, Cmat, cvec, members, item_inputs, out, B);
}